// KAN_19859928777111
// MI455X (gfx1250) — compile-verified
//
#include <hip/hip_runtime.h>

typedef __attribute__((ext_vector_type(2))) float v2f;
typedef __attribute__((ext_vector_type(8))) float v8f;

// Problem sizes (fixed by the reference)
constexpr int Bb = 64;       // batch
constexpr int Ii = 16384;    // input dim
constexpr int Oo = 1024;     // output dim
constexpr int Kk = 5;        // num basis MLPs
constexpr int Hh = 16;       // basis hidden
constexpr long KT = (long)Ii * Kk;   // flattened per-o row length of W (81920 floats)

constexpr int NCHUNK = 64;           // split-K chunks over I
constexpr int CHUNK  = Ii / NCHUNK;  // 256 i-values per wave

// ---------------------------------------------------------------------------
// Kernel 0: zero the output (harness poisons d_out; we accumulate atomically)
// ---------------------------------------------------------------------------
__global__ void kan_zero_kernel(float* __restrict__ p, int n) {
    int i = blockIdx.x * blockDim.x + threadIdx.x;
    if (i < n) p[i] = 0.0f;
}

// ---------------------------------------------------------------------------
// Kernel 1: basis[b,i] = w2_k . silu(x[b,i]*w1_k + b1_k) + b2_k,  k = i % 5
// Output layout: pair-interleaved transpose  basisP[(i/2)*128 + b*2 + (i&1)]
// so kernel 2 can fetch the (K, K+1) B-fragment pair as one aligned float2.
// Grid: Ii/64 blocks of 256 threads; each block does a 64i x 64b tile with an
// LDS transpose so both the x reads and the basisP writes are coalesced.
// ---------------------------------------------------------------------------
__global__ __launch_bounds__(256)
void kan_basis_kernel(const float* __restrict__ x,
                      const float* __restrict__ w1, const float* __restrict__ b1,
                      const float* __restrict__ w2, const float* __restrict__ b2,
                      float* __restrict__ basisP) {
    __shared__ float sW1[Kk * Hh], sB1[Kk * Hh], sW2[Kk * Hh], sB2[Kk];
    __shared__ float tile[64 * 65];            // [i_local][b_local], padded
    const int t = threadIdx.x;
    if (t < Kk * Hh) { sW1[t] = w1[t]; sB1[t] = b1[t]; sW2[t] = w2[t]; }
    if (t < Kk) sB2[t] = b2[t];
    __syncthreads();

    const int i0 = blockIdx.x * 64;
    #pragma unroll 4
    for (int rep = 0; rep < 16; ++rep) {
        int idx = t + rep * 256;               // 0..4095
        int il  = idx & 63;                    // i within tile (fast -> coalesced x)
        int bl  = idx >> 6;                    // batch row
        float xv = x[(long)bl * Ii + i0 + il];
        int  k   = (i0 + il) % Kk;
        float acc = sB2[k];
        #pragma unroll
        for (int h = 0; h < Hh; ++h) {
            float z = fmaf(xv, sW1[k * Hh + h], sB1[k * Hh + h]);
            float s = z * (1.0f / (1.0f + __expf(-z)));   // silu
            acc = fmaf(s, sW2[k * Hh + h], acc);
        }
        tile[il * 65 + bl] = acc;
    }
    __syncthreads();

    // Write out: thread handles (pair, b); consecutive lanes -> consecutive float2
    float2* outp = (float2*)basisP;            // [Ii/2][64] of float2
    #pragma unroll 4
    for (int rep = 0; rep < 8; ++rep) {
        int idx = t + rep * 256;               // 0..2047 = 32 pairs x 64 b
        int pl  = idx >> 6;                    // pair 0..31
        int bl  = idx & 63;
        float2 v;
        v.x = tile[(2 * pl    ) * 65 + bl];
        v.y = tile[(2 * pl + 1) * 65 + bl];
        outp[((long)(i0 >> 1) + pl) * 64 + bl] = v;
    }
}

// ---------------------------------------------------------------------------
// Kernel 2: out[b,o] += sum_i Wsum[o,i] * basis[b,i] via v_wmma_f32_16x16x4_f32
// GEMM view: M = o (1024), N = b (64), K = i (16384).
// A fragment (16x4 f32): lane<16 holds M=lane, K={0,1}; lane>=16 M=lane-16,
// K={2,3}.  Wsum is formed on the fly: each lane streams 10 consecutive
// floats of W (2 i-values x 5 k) as 5 aligned float2 loads + VALU adds.
// Each wave owns one o-tile and all 4 batch-tiles; split-K over 64 chunks.
// Block = 8 waves sharing one o-tile; LDS ds_add_f32 reduction, then one
// global_atomic_add_f32 pass.
// Grid: 512 blocks x 256 threads (64 o-tiles x 8 chunk-groups x 8 waves).
// ---------------------------------------------------------------------------
__global__ __launch_bounds__(256)
void kan_wmma_kernel(const float* __restrict__ Wm,
                     const float* __restrict__ basisP,
                     float* __restrict__ out) {
    __shared__ float red[4 * 16 * 16];         // [btile][M][N] f32 accumulators
    const int tid = threadIdx.x;
    #pragma unroll
    for (int e = tid; e < 1024; e += 256) red[e] = 0.0f;
    __syncthreads();

    const int lane = tid & 31;
    const int wave = tid >> 5;                 // 0..7
    const int mt   = blockIdx.x & 63;          // o-tile
    const int cg   = blockIdx.x >> 6;          // chunk group 0..7
    const int ch   = cg * 8 + wave;            // chunk 0..63
    const int m0   = mt * 16;
    const int kbase = ch * CHUNK;

    const int nlo  = lane & 15;
    const int koff = (lane >> 4) << 1;         // 0 (lanes 0-15) or 2 (lanes 16-31)
    const int row  = m0 + nlo;                 // this lane's o-row for A

    const float2* Ap = (const float2*)(Wm + (long)row * KT);  // row*KT*4 bytes, 8B aligned
    const float2* Bp = (const float2*)basisP;                 // [Ii/2][64]

    v8f acc0 = {0,0,0,0,0,0,0,0};
    v8f acc1 = {0,0,0,0,0,0,0,0};
    v8f acc2 = {0,0,0,0,0,0,0,0};
    v8f acc3 = {0,0,0,0,0,0,0,0};

    #pragma unroll 2
    for (int kk = 0; kk < CHUNK; kk += 4) {
        const int ka = kbase + kk + koff;      // even i index this lane covers
        // ---- A fragment: Wsum[row][ka], Wsum[row][ka+1] = sums of 5 W values
        const long f2 = ((long)ka * 5) >> 1;   // float2 index (5*ka is even)
        float2 p0 = Ap[f2 + 0];
        float2 p1 = Ap[f2 + 1];
        float2 p2 = Ap[f2 + 2];
        float2 p3 = Ap[f2 + 3];
        float2 p4 = Ap[f2 + 4];
        v2f a;
        a.x = ((p0.x + p0.y) + (p1.x + p1.y)) + p2.x;
        a.y = ((p2.y + p3.x) + (p3.y + p4.x)) + p4.y;
        // ---- B fragments: basis[(ka, ka+1)][b] pairs, one float2 per b-tile
        const long bp = ((long)(ka >> 1)) * 64;
        float2 q0 = Bp[bp +  0 + nlo];
        float2 q1 = Bp[bp + 16 + nlo];
        float2 q2 = Bp[bp + 32 + nlo];
        float2 q3 = Bp[bp + 48 + nlo];
        v2f vb0; vb0.x = q0.x; vb0.y = q0.y;
        v2f vb1; vb1.x = q1.x; vb1.y = q1.y;
        v2f vb2; vb2.x = q2.x; vb2.y = q2.y;
        v2f vb3; vb3.x = q3.x; vb3.y = q3.y;

        acc0 = __builtin_amdgcn_wmma_f32_16x16x4_f32(false, a, false, vb0, (short)0, acc0, false, false);
        acc1 = __builtin_amdgcn_wmma_f32_16x16x4_f32(false, a, false, vb1, (short)0, acc1, false, false);
        acc2 = __builtin_amdgcn_wmma_f32_16x16x4_f32(false, a, false, vb2, (short)0, acc2, false, false);
        acc3 = __builtin_amdgcn_wmma_f32_16x16x4_f32(false, a, false, vb3, (short)0, acc3, false, false);
    }

    // ---- reduce the 8 waves' partial C tiles in LDS (ds_add_f32)
    // D layout: VGPR r -> lanes 0-15: (M=r, N=lane); lanes 16-31: (M=r+8, N=lane-16)
    const int Mbase = (lane >> 4) << 3;
    #pragma unroll
    for (int r = 0; r < 8; ++r) {
        const int M = r + Mbase;
        atomicAdd(&red[0 * 256 + M * 16 + nlo], acc0[r]);
        atomicAdd(&red[1 * 256 + M * 16 + nlo], acc1[r]);
        atomicAdd(&red[2 * 256 + M * 16 + nlo], acc2[r]);
        atomicAdd(&red[3 * 256 + M * 16 + nlo], acc3[r]);
    }
    __syncthreads();

    // ---- one global atomic pass per block: out[b, o] with b = bt*16+N, o = m0+M
    #pragma unroll
    for (int e = tid; e < 1024; e += 256) {
        const int bt = e >> 8;
        const int M  = (e >> 4) & 15;
        const int N  = e & 15;
        atomicAdd(&out[(long)(bt * 16 + N) * Oo + m0 + M], red[e]);
    }
}

// ---------------------------------------------------------------------------
// Launch: zero out -> basis precompute (4 MB ws) -> fused Wsum+WMMA GEMM
// ---------------------------------------------------------------------------
extern "C" void kernel_launch(void* const* d_in, const int* in_sizes, int n_in,
                              void* d_out, int out_size, void* d_ws, size_t ws_size,
                              hipStream_t stream) {
    const float* x  = (const float*)d_in[0];
    const float* w1 = (const float*)d_in[1];
    const float* b1 = (const float*)d_in[2];
    const float* w2 = (const float*)d_in[3];
    const float* b2 = (const float*)d_in[4];
    const float* Wm = (const float*)d_in[5];
    float* out    = (float*)d_out;
    float* basisP = (float*)d_ws;              // needs Ii*Bb*4 = 4 MB

    const int nOut = Bb * Oo;                  // 65536
    kan_zero_kernel<<<(nOut + 255) / 256, 256, 0, stream>>>(out, nOut);
    kan_basis_kernel<<<Ii / 64, 256, 0, stream>>>(x, w1, b1, w2, b2, basisP);
    kan_wmma_kernel<<<(Oo / 16) * 8, 256, 0, stream>>>(Wm, basisP, out);
}